// DGAGNNLayer_3736621547759
// MI455X (gfx1250) — compile-verified
//
#include <hip/hip_runtime.h>
#include <hip/hip_bf16.h>

typedef __attribute__((ext_vector_type(2))) float v2f;
typedef __attribute__((ext_vector_type(4))) float v4f;
typedef __attribute__((ext_vector_type(8))) float v8f;

#define D 64

// ---------------------------------------------------------------------------
// Stage 0: zero the per-group aggregation buffers in workspace.
// ---------------------------------------------------------------------------
__global__ void zero_ws_kernel(v4f* __restrict__ p, long n4) {
    long i = (long)blockIdx.x * blockDim.x + threadIdx.x;
    if (i < n4) {
        v4f z = {0.f, 0.f, 0.f, 0.f};
        p[i] = z;
    }
}

// ---------------------------------------------------------------------------
// Stage 1: edge scatter. One wave32 per edge: lane l handles dims {2l, 2l+1}.
//   agg[group[src]][dst][:] += h[src][:]
// Edge metadata is wave-uniform -> scalar loads; h gather is a coalesced
// 256B row read; scatter via global_atomic_add_f32 (L2-resident).
// ---------------------------------------------------------------------------
__global__ void edge_scatter_kernel(const float* __restrict__ h,
                                    const int* __restrict__ src,
                                    const int* __restrict__ dst,
                                    const int* __restrict__ glab,
                                    float* __restrict__ agg,
                                    int E, int N) {
    int wave = (int)((blockIdx.x * (long)blockDim.x + threadIdx.x) >> 5);
    int lane = threadIdx.x & 31;
    if (wave >= E) return;

    int s = src[wave];
    int d = dst[wave];
    int g = glab[s];

    v2f hv = *(const v2f*)(h + (long)s * D + lane * 2);
    float* base = agg + ((long)g * N + d) * D + lane * 2;
    atomicAdd(base + 0, hv[0]);
    atomicAdd(base + 1, hv[1]);
}

// ---------------------------------------------------------------------------
// Stage 2: fused fp32 WMMA GEMM:
//   out = h @ Wself^T + agg0 @ W0^T + agg1 @ W1^T
// One wave per 16x16 output tile, K=64 swept as 16 x V_WMMA_F32_16X16X4_F32.
//
// 32-bit A (16x4, MxK) layout: lanes 0-15 -> M=lane, VGPR{0,1} = K{0,1};
//                              lanes 16-31 -> M=lane-16, VGPR{0,1} = K{2,3}.
// 32-bit B (4x16, KxN) layout: lanes 0-15 -> N=lane, VGPR{0,1} = K{0,1};
//                              lanes 16-31 -> N=lane-16, VGPR{0,1} = K{2,3}.
// B[k][n] = W[n][k] since the reference multiplies by W^T (W is [out,in]).
// ---------------------------------------------------------------------------
__device__ __forceinline__ v8f wmma_acc_mat(const float* __restrict__ A,
                                            const float* __restrict__ W,
                                            int rowBase, int jBase,
                                            int lane, v8f acc) {
    const int m    = lane & 15;
    const int koff = (lane >> 4) * 2;
    const float* arow = A + (long)(rowBase + m) * D;   // A[m][*]
    const float* wrow = W + (long)(jBase + m) * D;     // W[n][*] -> B[*][n]
#pragma unroll
    for (int ks = 0; ks < D / 4; ++ks) {
        v2f a = *(const v2f*)(arow + ks * 4 + koff);
        v2f b = *(const v2f*)(wrow + ks * 4 + koff);
        acc = __builtin_amdgcn_wmma_f32_16x16x4_f32(
            /*neg_a=*/false, a, /*neg_b=*/false, b,
            /*c_mod=*/(short)0, acc, /*reuse_a=*/false, /*reuse_b=*/false);
    }
    return acc;
}

__global__ void gemm_wmma_kernel(const float* __restrict__ h,
                                 const float* __restrict__ Wself,
                                 const float* __restrict__ Wg,
                                 const float* __restrict__ agg,
                                 float* __restrict__ out,
                                 int nTilesM, int N) {
    int wave = (int)((blockIdx.x * (long)blockDim.x + threadIdx.x) >> 5);
    int lane = threadIdx.x & 31;
    int tileM = wave >> 2;   // 16-row tile index
    int tileJ = wave & 3;    // 16-col tile index (D=64 -> 4 tiles)
    if (tileM >= nTilesM) return;   // wave-uniform; EXEC stays all-ones

    int rowBase = tileM * 16;
    int jBase   = tileJ * 16;

    v8f acc = {};
    acc = wmma_acc_mat(h,                 Wself,        rowBase, jBase, lane, acc);
    acc = wmma_acc_mat(agg,               Wg,           rowBase, jBase, lane, acc);
    acc = wmma_acc_mat(agg + (long)N * D, Wg + D * D,   rowBase, jBase, lane, acc);

    // 32-bit C/D 16x16 layout: lanes 0-15 -> rows 0-7 in VGPRs 0-7 (N=lane);
    // lanes 16-31 -> rows 8-15.
    int m0  = (lane >> 4) * 8;
    int col = lane & 15;
#pragma unroll
    for (int r = 0; r < 8; ++r) {
        out[(long)(rowBase + m0 + r) * D + jBase + col] = acc[r];
    }
}

// ---------------------------------------------------------------------------
// Launch
// ---------------------------------------------------------------------------
extern "C" void kernel_launch(void* const* d_in, const int* in_sizes, int n_in,
                              void* d_out, int out_size, void* d_ws, size_t ws_size,
                              hipStream_t stream) {
    const float* h     = (const float*)d_in[0];
    const float* Wself = (const float*)d_in[1];
    const float* Wg    = (const float*)d_in[2];
    const int*   src   = (const int*)d_in[3];
    const int*   dst   = (const int*)d_in[4];
    const int*   glab  = (const int*)d_in[5];
    float*       out   = (float*)d_out;

    const int N = in_sizes[0] / D;        // 100000
    const int E = in_sizes[3];            // 1600000
    const int G = in_sizes[2] / (D * D);  // 2

    float* agg = (float*)d_ws;            // [G][N][D] fp32
    long aggElems = (long)G * N * D;

    // Stage 0: zero aggregation scratch (every call; harness poisons ws).
    long n4 = aggElems / 4;
    int zb = (int)((n4 + 255) / 256);
    zero_ws_kernel<<<zb, 256, 0, stream>>>((v4f*)agg, n4);

    // Stage 1: edge gather + atomic scatter-add into per-group buffers.
    int eb = (E + 7) / 8;                 // 8 waves (edges) per 256-thread block
    edge_scatter_kernel<<<eb, 256, 0, stream>>>(h, src, dst, glab, agg, E, N);

    // Stage 2: fused WMMA GEMM -> d_out.
    int nTilesM = N / 16;                 // 6250 (exact)
    int waves   = nTilesM * 4;            // 25000 tiles
    int gb      = (waves + 7) / 8;
    gemm_wmma_kernel<<<gb, 256, 0, stream>>>(h, Wself, Wg, agg, out, nTilesM, N);
}